// ICPEstimator_39479339385262
// MI455X (gfx1250) — compile-verified
//
#include <hip/hip_runtime.h>
#include <math.h>

// ---------------- problem constants (match reference) ----------------
static constexpr int kH = 768;
static constexpr int kW = 1024;
static constexpr int kIters = 10;
static constexpr float kDistThr = 100.0f / 15.0f;
static constexpr float kNormalThr = 0.34f;
static constexpr float kDamp = 1e-9f;

typedef float v2f __attribute__((ext_vector_type(2)));
typedef float v8f __attribute__((ext_vector_type(8)));

// ---------------- workspace layout (floats) ----------------
// [0..5]   x (6-vector pose parameters)
// [8..19]  pose: R row-major (9) then t (3)
// [32 ...] per-iteration 16x16 f32 accumulators (256 floats each)
static constexpr int kXOff = 0;
static constexpr int kPoseOff = 8;
static constexpr int kAccOff = 32;
static constexpr int kAccSize = 256;
static constexpr int kWsFloats = kAccOff + kIters * kAccSize;

__global__ void icp_init(float* ws) {
  int i = blockIdx.x * blockDim.x + threadIdx.x;
  if (i < kWsFloats) {
    float v = 0.f;
    // identity rotation R00,R11,R22 at pose offsets 0,4,8
    if (i == kPoseOff + 0 || i == kPoseOff + 4 || i == kPoseOff + 8) v = 1.f;
    ws[i] = v;
  }
}

__device__ __forceinline__ void backproject(const float* Kinv,
                                            const float* __restrict__ depth,
                                            int u, int v, float out[3]) {
  float d = depth[v * kW + u];
  float uf = (float)u, vf = (float)v;
  out[0] = (Kinv[0] * uf + Kinv[1] * vf + Kinv[2]) * d;
  out[1] = (Kinv[3] * uf + Kinv[4] * vf + Kinv[5]) * d;
  out[2] = (Kinv[6] * uf + Kinv[7] * vf + Kinv[8]) * d;
}

#define BLOCK 256
#define WAVES_PER_BLOCK (BLOCK / 32)

// Per-point residual/Jacobian, reduced via V_WMMA_F32_16X16X4_F32:
// D += A(16x4) x B(4x16) where A^T == B == 4 extended-Jacobian rows.
__global__ void __launch_bounds__(BLOCK) icp_accum(
    const float* __restrict__ pts, const float* __restrict__ nrms,
    const float* __restrict__ depth, const float* __restrict__ Km,
    const float* __restrict__ pose, float* __restrict__ gacc, int N) {
  __shared__ float jbuf[WAVES_PER_BLOCK][8 * 32];  // [wave][col*32 + point]
  __shared__ float accbuf[256];

  const int tid = threadIdx.x;
  const int lane = tid & 31;
  const int wv = tid >> 5;
  accbuf[tid] = 0.f;
  __syncthreads();

  // wave-uniform loads: pose, K, K^-1 (emitted as scalar loads)
  float R[9], t[3], K9[9], Kinv[9];
#pragma unroll
  for (int i = 0; i < 9; i++) R[i] = pose[i];
#pragma unroll
  for (int i = 0; i < 3; i++) t[i] = pose[9 + i];
#pragma unroll
  for (int i = 0; i < 9; i++) K9[i] = Km[i];
  {
    float a = K9[0], b = K9[1], c = K9[2], d = K9[3], e = K9[4], f = K9[5],
          g = K9[6], h = K9[7], ii = K9[8];
    float A0 = e * ii - f * h, A1 = c * h - b * ii, A2 = b * f - c * e;
    float A3 = f * g - d * ii, A4 = a * ii - c * g, A5 = c * d - a * f;
    float A6 = d * h - e * g, A7 = b * g - a * h, A8 = a * e - b * d;
    float inv = 1.f / (a * A0 + b * A3 + c * A6);
    Kinv[0] = A0 * inv; Kinv[1] = A1 * inv; Kinv[2] = A2 * inv;
    Kinv[3] = A3 * inv; Kinv[4] = A4 * inv; Kinv[5] = A5 * inv;
    Kinv[6] = A6 * inv; Kinv[7] = A7 * inv; Kinv[8] = A8 * inv;
  }

  const int nWaves = (gridDim.x * BLOCK) >> 5;
  const int gwave = (blockIdx.x * BLOCK + tid) >> 5;
  const int totalBatches = (N + 31) / 32;
  const int hh = lane >> 4;  // half-wave: selects points {0,1} vs {2,3}
  const int cc = lane & 15;  // matrix row/col index this lane owns

  v8f acc = {0.f, 0.f, 0.f, 0.f, 0.f, 0.f, 0.f, 0.f};

  for (int batch = gwave; batch < totalBatches; batch += nWaves) {
    const int pidx = batch * 32 + lane;
    const int li = pidx < N ? pidx : 0;

    // ---- per-point pipeline ----
    float px = pts[3 * li + 0], py = pts[3 * li + 1], pz = pts[3 * li + 2];
    float nx = nrms[3 * li + 0], ny = nrms[3 * li + 1], nz = nrms[3 * li + 2];
    float dx = px - t[0], dy = py - t[1], dz = pz - t[2];
    // pc = R^T (p - t), nc = R^T n
    float pcx = R[0] * dx + R[3] * dy + R[6] * dz;
    float pcy = R[1] * dx + R[4] * dy + R[7] * dz;
    float pcz = R[2] * dx + R[5] * dy + R[8] * dz;
    float ncx = R[0] * nx + R[3] * ny + R[6] * nz;
    float ncy = R[1] * nx + R[4] * ny + R[7] * nz;
    float ncz = R[2] * nx + R[5] * ny + R[8] * nz;
    // project
    float uw = K9[0] * pcx + K9[1] * pcy + K9[2] * pcz;
    float vw = K9[3] * pcx + K9[4] * pcy + K9[5] * pcz;
    float zw = K9[6] * pcx + K9[7] * pcy + K9[8] * pcz;
    float zs = fabsf(zw) > 1e-8f ? zw : 1e-8f;
    float u = uw / zs, v = vw / zs;
    bool valid = (u > 0.f) && (u < (float)kW) && (v > 0.f) &&
                 (v < (float)kH) && (zw > 1e-6f) && (pidx < N);
    int iu = (int)fminf(fmaxf(u, 0.f), (float)(kW - 1));
    int iv = (int)fminf(fmaxf(v, 0.f), (float)(kH - 1));

    // reference-grid point + grid normal, recomputed from L2-resident depth
    float rg[3];
    backproject(Kinv, depth, iu, iv, rg);
    int u0 = iu > 0 ? iu - 1 : 0;
    int u1 = iu < kW - 1 ? iu + 1 : kW - 1;
    float su = (iu > 0 && iu < kW - 1) ? 0.5f : 1.f;
    int v0 = iv > 0 ? iv - 1 : 0;
    int v1 = iv < kH - 1 ? iv + 1 : kH - 1;
    float sv = (iv > 0 && iv < kH - 1) ? 0.5f : 1.f;
    float pu1[3], pu0[3], pv1[3], pv0[3];
    backproject(Kinv, depth, u1, iv, pu1);
    backproject(Kinv, depth, u0, iv, pu0);
    backproject(Kinv, depth, iu, v1, pv1);
    backproject(Kinv, depth, iu, v0, pv0);
    float dux = (pu1[0] - pu0[0]) * su, duy = (pu1[1] - pu0[1]) * su,
          duz = (pu1[2] - pu0[2]) * su;
    float dvx = (pv1[0] - pv0[0]) * sv, dvy = (pv1[1] - pv0[1]) * sv,
          dvz = (pv1[2] - pv0[2]) * sv;
    float nrx = duy * dvz - duz * dvy;
    float nry = duz * dvx - dux * dvz;
    float nrz = dux * dvy - duy * dvx;
    float ninv = 1.f / sqrtf(nrx * nrx + nry * nry + nrz * nrz + 1e-12f);
    nrx *= ninv; nry *= ninv; nrz *= ninv;

    float ddx = rg[0] - pcx, ddy = rg[1] - pcy, ddz = rg[2] - pcz;
    float dist = sqrtf(ddx * ddx + ddy * ddy + ddz * ddz + 1e-12f);
    float ndot = nrx * ncx + nry * ncy + nrz * ncz;
    valid = valid && (dist < kDistThr) && (ndot > kNormalThr);
    float w = valid ? 1.f : 0.f;

    // ref point in world frame: rw = R*rg + t
    float rwx = R[0] * rg[0] + R[1] * rg[1] + R[2] * rg[2] + t[0];
    float rwy = R[3] * rg[0] + R[4] * rg[1] + R[5] * rg[2] + t[1];
    float rwz = R[6] * rg[0] + R[7] * rg[1] + R[8] * rg[2] + t[2];
    float r = (nx * (rwx - px) + ny * (rwy - py) + nz * (rwz - pz)) * w;

    float jrow[8];
    jrow[0] = (rwy * nz - rwz * ny) * w;  // cross(rw, n) . x
    jrow[1] = (rwz * nx - rwx * nz) * w;
    jrow[2] = (rwx * ny - rwy * nx) * w;
    jrow[3] = nx * w;
    jrow[4] = ny * w;
    jrow[5] = nz * w;
    jrow[6] = r;
    jrow[7] = w;

    // stage into LDS: [col][point] for the WMMA operand transpose.
    // Same-wave LDS ops are in-order (DScnt); only this wave reads its region.
#pragma unroll
    for (int j = 0; j < 8; j++) jbuf[wv][j * 32 + lane] = jrow[j];
    __builtin_amdgcn_wave_barrier();

    // 8 WMMA steps x 4 points: D(16x16) += Jext^T(16x4) * Jext(4x16).
    // Lane (hh,cc): A/B operand = {Jext[p0][cc], Jext[p0+1][cc]},
    // p0 = step*4 + 2*hh  (A-layout: K split low/high half-wave; same for B).
#pragma unroll
    for (int step = 0; step < 8; step++) {
      int p0 = step * 4 + 2 * hh;
      v2f ab;
      ab.x = (cc < 8) ? jbuf[wv][cc * 32 + p0] : 0.f;
      ab.y = (cc < 8) ? jbuf[wv][cc * 32 + p0 + 1] : 0.f;
      acc = __builtin_amdgcn_wmma_f32_16x16x4_f32(
          false, ab, false, ab, (short)0, acc, false, false);
    }
    __builtin_amdgcn_wave_barrier();
  }

  // C/D layout: lanes 0-15 vgpr j -> (M=j, N=lane); lanes 16-31 -> (M=j+8).
#pragma unroll
  for (int j = 0; j < 8; j++) {
    atomicAdd(&accbuf[(j + 8 * hh) * 16 + cc], acc[j]);
  }
  __syncthreads();
  atomicAdd(&gacc[tid], accbuf[tid]);
}

// One-thread Gauss-Newton step: read 16x16 accumulator, solve 6x6, update x,
// recompute pose via SO3 exp; on the last iteration emit T (4x4) + cost.
__global__ void icp_solve(float* ws, float* out, int iter) {
  if (threadIdx.x != 0 || blockIdx.x != 0) return;
  const float* acc = ws + kAccOff + iter * kAccSize;

  float A[6][7];
#pragma unroll
  for (int i = 0; i < 6; i++) {
#pragma unroll
    for (int j = 0; j < 6; j++) A[i][j] = acc[i * 16 + j];
    A[i][i] += kDamp;
    A[i][6] = acc[i * 16 + 6];  // J^T r
  }
  float cost = acc[6 * 16 + 6] / fmaxf(acc[7 * 16 + 7], 1.f);

  // Gaussian elimination with partial pivoting
  for (int k = 0; k < 6; k++) {
    int piv = k;
    float mx = fabsf(A[k][k]);
    for (int i = k + 1; i < 6; i++) {
      float a = fabsf(A[i][k]);
      if (a > mx) { mx = a; piv = i; }
    }
    if (piv != k) {
      for (int j = k; j < 7; j++) {
        float tmp = A[k][j]; A[k][j] = A[piv][j]; A[piv][j] = tmp;
      }
    }
    float inv = 1.f / A[k][k];
    for (int i = k + 1; i < 6; i++) {
      float f = A[i][k] * inv;
      for (int j = k; j < 7; j++) A[i][j] -= f * A[k][j];
    }
  }
  float sol[6];
  for (int i = 5; i >= 0; i--) {
    float s = A[i][6];
    for (int j = i + 1; j < 6; j++) s -= A[i][j] * sol[j];
    sol[i] = s / A[i][i];
  }

  float x[6];
#pragma unroll
  for (int i = 0; i < 6; i++) {
    x[i] = ws[kXOff + i] - sol[i];
    ws[kXOff + i] = x[i];
  }

  // SO3 exp: R = I + sinc(th)*W + (1-cos th)/th2 * W^2,  W^2 = w w^T - th2 I
  float wx = x[0], wy = x[1], wz = x[2];
  float th2 = wx * wx + wy * wy + wz * wz;
  float th = sqrtf(th2 + 1e-12f);
  float sA = sinf(th) / th;
  float sB = (1.f - cosf(th)) / (th2 + 1e-12f);
  float Rm[9];
  Rm[0] = 1.f + sB * (wx * wx - th2);
  Rm[1] = -sA * wz + sB * wx * wy;
  Rm[2] =  sA * wy + sB * wx * wz;
  Rm[3] =  sA * wz + sB * wy * wx;
  Rm[4] = 1.f + sB * (wy * wy - th2);
  Rm[5] = -sA * wx + sB * wy * wz;
  Rm[6] = -sA * wy + sB * wz * wx;
  Rm[7] =  sA * wx + sB * wz * wy;
  Rm[8] = 1.f + sB * (wz * wz - th2);

#pragma unroll
  for (int i = 0; i < 9; i++) ws[kPoseOff + i] = Rm[i];
#pragma unroll
  for (int i = 0; i < 3; i++) ws[kPoseOff + 9 + i] = x[3 + i];

  if (iter == kIters - 1) {
    // T row-major, then cost
    out[0] = Rm[0]; out[1] = Rm[1]; out[2] = Rm[2]; out[3] = x[3];
    out[4] = Rm[3]; out[5] = Rm[4]; out[6] = Rm[5]; out[7] = x[4];
    out[8] = Rm[6]; out[9] = Rm[7]; out[10] = Rm[8]; out[11] = x[5];
    out[12] = 0.f; out[13] = 0.f; out[14] = 0.f; out[15] = 1.f;
    out[16] = cost;
  }
}

extern "C" void kernel_launch(void* const* d_in, const int* in_sizes, int n_in,
                              void* d_out, int out_size, void* d_ws,
                              size_t ws_size, hipStream_t stream) {
  const float* ref_depth = (const float*)d_in[0];
  const float* target_pts = (const float*)d_in[1];
  const float* target_normals = (const float*)d_in[2];
  const float* Km = (const float*)d_in[3];
  float* ws = (float*)d_ws;
  float* out = (float*)d_out;
  const int N = in_sizes[1] / 3;

  icp_init<<<(kWsFloats + 255) / 256, 256, 0, stream>>>(ws);
  for (int it = 0; it < kIters; it++) {
    icp_accum<<<256, BLOCK, 0, stream>>>(target_pts, target_normals, ref_depth,
                                         Km, ws + kPoseOff,
                                         ws + kAccOff + it * kAccSize, N);
    icp_solve<<<1, 32, 0, stream>>>(ws, out, it);
  }
}